// MultiScaleLoRA_40398462386258
// MI455X (gfx1250) — compile-verified
//
#include <hip/hip_runtime.h>
#include <hip/hip_bf16.h>

typedef __attribute__((ext_vector_type(2))) float v2f;
typedef __attribute__((ext_vector_type(8))) float v8f;
typedef __attribute__((ext_vector_type(4))) unsigned int v4u;
typedef __attribute__((ext_vector_type(8))) int v8i;
typedef __attribute__((ext_vector_type(4))) int v4i;

#define M_TOTAL    16384
#define KDIM       4096
#define RDIM       96
#define NDIM       4096
#define LORA_SCALE (1.0f / 32.0f)

#define BLOCK_ROWS 128
#define NTHREADS   256
#define KCHUNK     64
#define NCHUNK     256
#define NKCHUNKS   (KDIM / KCHUNK)

// LDS strides in floats, padded so WMMA operand reads are bank-conflict-free
#define HSTRIDE  100  // h rows:   96 + 4
#define XSTRIDE  68   // x rows:   64 + 4  (TDM pad: 4 dwords per 64 dwords)
#define ATSTRIDE 68   // A^T rows: 64 + 4
#define BTSTRIDE 102  // B^T rows: 96 + 6

#define XBUF_FLOATS  (BLOCK_ROWS * XSTRIDE)   // 8704
#define ATBUF_FLOATS (RDIM * ATSTRIDE)        // 6528
#define HBUF_FLOATS  (BLOCK_ROWS * HSTRIDE)   // 12800
#define BTBUF_FLOATS (NCHUNK * BTSTRIDE)      // 26112
// LDS layout (floats). achkT double-buffer first: bytes 0..52224 are all
// reachable with the 16-bit DS immediate offset, so the 6 B-operand loads
// per k-step need no per-load address adds.
#define ATBUF0_OFF 0
#define XBUF0_OFF  (2 * ATBUF_FLOATS)              // 13056
#define HBUF_OFF   (XBUF0_OFF + 2 * XBUF_FLOATS)   // 30464
#define SMEM_FLOATS (HBUF_OFF + HBUF_FLOATS)       // 43264
#define SMEM_BYTES  (SMEM_FLOATS * 4)              // 173056 bytes < 320KB/WGP
// stage 2: bchkT (26112 floats) aliases floats 0..26112 (achkT+xbuf region,
// dead once stage 1 finishes and ha is register-resident)

static __device__ __forceinline__ v8f wmma_f32_k4(v2f a, v2f b, v8f c) {
  return __builtin_amdgcn_wmma_f32_16x16x4_f32(false, a, false, b, (short)0, c,
                                               false, false);
}

// TDM: DMA one x chunk (BLOCK_ROWS x KCHUNK fp32 2D tile, row stride KDIM)
// straight into LDS, padding each 64-dword row with 4 dwords -> XSTRIDE=68.
// D# per CDNA5 ISA §8.3-8.6. Issued by one wave; tracked with TENSORcnt.
static __device__ __forceinline__ void tdm_load_x_chunk(const float* gsrc,
                                                        unsigned ldsByteAddr) {
  const unsigned long long ga = (unsigned long long)(size_t)gsrc;
  v4u g0;
  g0[0] = 1u;                                   // count=1, user descriptor
  g0[1] = ldsByteAddr;                          // lds_addr
  g0[2] = (unsigned)ga;                         // global_addr[31:0]
  g0[3] = (unsigned)((ga >> 32) & 0x01FFFFFFull) | (2u << 30);  // type=2
  v8i g1;
  g1[0] = (2 << 16)        // data_size = 4 bytes
        | (1 << 20)        // pad_enable
        | (5 << 22)        // pad_interval: every 64 dwords
        | (3 << 25);       // pad_amount: 4 dwords
  g1[1] = (KCHUNK & 0xFFFF) << 16;                        // tensor_dim0 lo16
  g1[2] = (BLOCK_ROWS & 0xFFFF) << 16;                    // tensor_dim1 lo16
  g1[3] = KCHUNK << 16;                                   // tile_dim0
  g1[4] = BLOCK_ROWS;                                     // tile_dim1 (dim2=0)
  g1[5] = KDIM;                                           // dim0 stride lo32
  g1[6] = 0;
  g1[7] = 0;
  v4i g2 = {};
  v4i g3 = {};
  v8i g4 = {};
  __builtin_amdgcn_tensor_load_to_lds(g0, g1, g2, g3, g4, 0);
}

__global__ __launch_bounds__(NTHREADS)
void MultiScaleLoRA_fused_wmma(const float* __restrict__ x,
                               const float* __restrict__ Af,
                               const float* __restrict__ Am,
                               const float* __restrict__ As,
                               const float* __restrict__ Bf,
                               const float* __restrict__ Bm,
                               const float* __restrict__ Bs,
                               float* __restrict__ out) {
  extern __shared__ float smem[];
  float* achkT0 = smem;                  // 2 x [96][ATSTRIDE] (double buffer)
  float* xbuf0  = smem + XBUF0_OFF;      // 2 x [128][XSTRIDE] (double buffer)
  float* hbuf   = smem + HBUF_OFF;       // [128][HSTRIDE]
  float* bchkT  = smem;                  // [256][BTSTRIDE] (stage 2, aliases)

  const int tid     = threadIdx.x;
  const int lane    = tid & 31;
  const int half    = lane >> 4;
  const int ln16    = lane & 15;
  const int wave    = tid >> 5;
  const int wRow    = wave << 4;
  const int rowBase = blockIdx.x * BLOCK_ROWS;
  // wave-uniform scalar condition -> scalar branch (TDM ignores EXEC, so the
  // issuing region must be skipped by a *scalar* branch in waves 1..7)
  const bool isWave0 = (__builtin_amdgcn_readfirstlane(wave) == 0);

  v8f zero = {};
  v8f acc[6];
#pragma unroll
  for (int j = 0; j < 6; ++j) acc[j] = zero;

  // ---------------- Stage 1: h[128x96] = x_block[128x4096] * Acat[4096x96]
  // Software pipeline: TDM chunk i+1 + A-gather i+1 overlap compute of i.
  if (isWave0)
    tdm_load_x_chunk(x + (size_t)rowBase * KDIM, (unsigned)(size_t)xbuf0);
  for (int e = tid; e < KCHUNK * RDIM; e += NTHREADS) {  // A chunk 0
    const int kk = e / RDIM;
    const int r  = e - kk * RDIM;
    const float* src = (r < 32) ? Af : ((r < 64) ? Am : As);
    achkT0[r * ATSTRIDE + kk] = src[(size_t)kk * 32 + (r & 31)];
  }

  for (int i = 0; i < NKCHUNKS; ++i) {
    const int cur = i & 1;
    const float* xbuf  = xbuf0 + cur * XBUF_FLOATS;
    const float* achkT = achkT0 + cur * ATBUF_FLOATS;
    __builtin_amdgcn_s_wait_tensorcnt(0);  // chunk i DMA complete (wave 0)
    __syncthreads();                       // visible to all; buf cur^1 free
    if (i + 1 < NKCHUNKS) {
      const int k0n = (i + 1) * KCHUNK;
      if (isWave0)
        tdm_load_x_chunk(
            x + (size_t)rowBase * KDIM + k0n,
            (unsigned)(size_t)(xbuf0 + (cur ^ 1) * XBUF_FLOATS));
      float* atn = achkT0 + (cur ^ 1) * ATBUF_FLOATS;
      for (int e = tid; e < KCHUNK * RDIM; e += NTHREADS) {
        const int kk = e / RDIM;
        const int r  = e - kk * RDIM;
        const float* src = (r < 32) ? Af : ((r < 64) ? Am : As);
        atn[r * ATSTRIDE + kk] = src[(size_t)(k0n + kk) * 32 + (r & 31)];
      }
    }
#pragma unroll 4
    for (int kk = 0; kk < KCHUNK / 4; ++kk) {
      const int arow = (kk << 2) + (half << 1);
      const v2f a = *(const v2f*)(xbuf + (wRow + ln16) * XSTRIDE + arow);
#pragma unroll
      for (int j = 0; j < 6; ++j) {
        const v2f b =
            *(const v2f*)(achkT + ((j << 4) + ln16) * ATSTRIDE + arow);
        acc[j] = wmma_f32_k4(a, b, acc[j]);
      }
    }
  }

  // spill h tile (C-layout) to LDS row-major, reload in A-operand layout
  // into registers once (this wave's h is private; no barrier needed).
#pragma unroll
  for (int j = 0; j < 6; ++j) {
#pragma unroll
    for (int v = 0; v < 8; ++v) {
      hbuf[(wRow + v + (half << 3)) * HSTRIDE + (j << 4) + ln16] = acc[j][v];
    }
  }
  v2f ha[RDIM / 4];
#pragma unroll
  for (int kk = 0; kk < RDIM / 4; ++kk) {
    ha[kk] = *(const v2f*)(hbuf + (wRow + ln16) * HSTRIDE + (kk << 2) +
                           (half << 1));
  }

  // ---------------- Stage 2: out[128x4096] = h[128x96] * Bcat[96x4096] * s
  for (int n0 = 0; n0 < NDIM; n0 += NCHUNK) {
    __syncthreads();  // guards stage-1 buffer reads & prior bchkT reads
    // B chunk: coalesced float4 global reads, transposed scalar LDS writes
    for (int e = tid; e < RDIM * (NCHUNK / 4); e += NTHREADS) {
      const int r  = e >> 6;
      const int c4 = (e & 63) << 2;
      const float* src = (r < 32) ? Bf : ((r < 64) ? Bm : Bs);
      const float4 v = *(const float4*)(src + (size_t)(r & 31) * NDIM + n0 + c4);
      bchkT[(c4 + 0) * BTSTRIDE + r] = v.x;
      bchkT[(c4 + 1) * BTSTRIDE + r] = v.y;
      bchkT[(c4 + 2) * BTSTRIDE + r] = v.z;
      bchkT[(c4 + 3) * BTSTRIDE + r] = v.w;
    }
    __syncthreads();
    for (int jj = 0; jj < NCHUNK / 16; ++jj) {
      v8f o = zero;
#pragma unroll 4
      for (int kk = 0; kk < RDIM / 4; ++kk) {
        const int arow = (kk << 2) + (half << 1);
        const v2f b =
            *(const v2f*)(bchkT + ((jj << 4) + ln16) * BTSTRIDE + arow);
        o = wmma_f32_k4(ha[kk], b, o);
      }
#pragma unroll
      for (int v = 0; v < 8; ++v) {
        const int row = rowBase + wRow + v + (half << 3);
        out[(size_t)row * NDIM + n0 + (jj << 4) + ln16] = o[v] * LORA_SCALE;
      }
    }
  }
}

extern "C" void kernel_launch(void* const* d_in, const int* in_sizes, int n_in,
                              void* d_out, int out_size, void* d_ws, size_t ws_size,
                              hipStream_t stream) {
  (void)in_sizes; (void)n_in; (void)d_ws; (void)ws_size; (void)out_size;
  const float* x  = (const float*)d_in[0];
  const float* Af = (const float*)d_in[1];
  const float* Bf = (const float*)d_in[2];
  const float* Am = (const float*)d_in[3];
  const float* Bm = (const float*)d_in[4];
  const float* As = (const float*)d_in[5];
  const float* Bs = (const float*)d_in[6];
  float* out = (float*)d_out;

  dim3 grid(M_TOTAL / BLOCK_ROWS);  // 128 blocks x 8 wave32 waves
  dim3 block(NTHREADS);
  MultiScaleLoRA_fused_wmma<<<grid, block, SMEM_BYTES, stream>>>(
      x, Af, Am, As, Bf, Bm, Bs, out);
}